// WaterNetModel_26414048870932
// MI455X (gfx1250) — compile-verified
//
#include <hip/hip_runtime.h>
#include <math.h>

// Problem constants (from reference: NT, NS, NH)
constexpr int NT = 1096;
constexpr int NS = 2048;
constexpr int NH = 16;

constexpr int BLOCK  = 64;               // 2 waves per workgroup
constexpr int SPB    = BLOCK / NH;       // 4 sites per block
constexpr int SPW    = 32 / NH;          // 2 sites per wave
constexpr int UNROLL = 8;                // time steps per body
constexpr int NBUF   = 16;               // two groups of 8 LDS buffers
constexpr size_t ROW = (size_t)NS * NH;  // elements per H/S time row

// Sum over the 16 lanes of each half-wave (bucket dimension).
// Preferred: v_permlane16_b32 xor-butterfly -- pure VALU, no DScnt waits.
// Fallback: ds_swizzle xor patterns (DS pipe + waits).
__device__ __forceinline__ float red16(float v) {
#if __has_builtin(__builtin_amdgcn_permlane16)
    int x;
    x = __float_as_int(v);
    v += __int_as_float(__builtin_amdgcn_permlane16(x, x, 0x67452301u, 0xEFCDAB89u, false, false)); // ^1
    x = __float_as_int(v);
    v += __int_as_float(__builtin_amdgcn_permlane16(x, x, 0x54761032u, 0xDCFE98BAu, false, false)); // ^2
    x = __float_as_int(v);
    v += __int_as_float(__builtin_amdgcn_permlane16(x, x, 0x32107654u, 0xBA98FEDCu, false, false)); // ^4
    x = __float_as_int(v);
    v += __int_as_float(__builtin_amdgcn_permlane16(x, x, 0xFEDCBA98u, 0x76543210u, false, false)); // ^8
#else
    v += __int_as_float(__builtin_amdgcn_ds_swizzle(__float_as_int(v), 0x041F));
    v += __int_as_float(__builtin_amdgcn_ds_swizzle(__float_as_int(v), 0x081F));
    v += __int_as_float(__builtin_amdgcn_ds_swizzle(__float_as_int(v), 0x101F));
    v += __int_as_float(__builtin_amdgcn_ds_swizzle(__float_as_int(v), 0x201F));
#endif
    return v;
}

__global__ __launch_bounds__(BLOCK)
void waternet_scan_kernel(const float* __restrict__ P,
                          const float* __restrict__ T,
                          const float* __restrict__ w_i,
                          const float* __restrict__ w_o,
                          const float* __restrict__ w_l,
                          const float* __restrict__ w_s,
                          float* __restrict__ Qout,
                          float* __restrict__ Hout,
                          float* __restrict__ Sout)
{
    // [buf][0=P,1=T][local site] : 16 buffers x 32 B = 512 B of LDS
    __shared__ float ldsPT[NBUF][2][SPB];

    const int tid    = threadIdx.x;
    const int b      = tid & (NH - 1);           // bucket 0..15
    const int sl     = tid >> 4;                 // local site 0..3
    const int site   = blockIdx.x * SPB + sl;
    const int lane   = tid & 31;
    const int waveSl = (tid >> 5) * SPW;         // local site base of this wave

    // ---- time-invariant gate factors (off the hot loop) ----
    const float wi = w_i[b], wo = w_o[b], wl = w_l[b], ws = w_s[b];
    const float melt = expf(ws) + 1.0f;              // exp(w_s) + 1
    const float gi   = 1.0f / (1.0f + expf(-wi));    // sigmoid(w_i)
    const float gl   = 1.0f / (1.0f + expf(-wl));    // sigmoid(w_l)
    const float e    = expf(wo);                     // softmax(w_o)
    const float a    = e / red16(e);

    // ---- async DMA setup: lane0 -> P pair, lane1 -> T pair (per wave) ----
    // One global_load_async_to_lds_b64 per wave per time step: lane0 copies
    // P[t, site0..site1], lane1 copies T[t, site0..site1] (8 B, 8 B aligned).
    const float* pfPtr = ((lane & 1) ? T : P)
                       + (size_t)blockIdx.x * SPB + waveSl;
    const unsigned ldsBase =
        (unsigned)(unsigned long long)(void*)&ldsPT[0][lane & 1][waveSl];

    auto issue = [&](int t, int buf) {
        if (lane < 2) {
            unsigned long long ga = (unsigned long long)(pfPtr + (size_t)t * NS);
            unsigned lo = ldsBase + (unsigned)(buf * (int)sizeof(ldsPT[0]));
            asm volatile("global_load_async_to_lds_b64 %0, %1, off"
                         :: "v"(lo), "v"(ga) : "memory");
        }
    };

    float s = 0.0f, h = 0.0f;                    // carried state (s0 = h0 = 0)

    // Prologue: stage t = 0..15 into buffers 0..15 (buf == t & 15)
#pragma unroll
    for (int j = 0; j < NBUF; ++j) issue(j, j);

    float* Qp = Qout;                            // bumped by UNROLL rows / body
    float* Hp = Hout;
    float* Sp = Sout;
    const size_t hsOff = (size_t)site * NH + b;

    // One body = 8 time steps from one LDS buffer group (all offsets static).
    auto body = [&](int k) {
        const int bb = k & 8;                    // buffer group base (0 or 8)
        float qs[UNROLL];
#pragma unroll
        for (int j = 0; j < UNROLL; ++j) {
            const float Pk = ldsPT[bb + j][0][sl];
            const float Tk = ldsPT[bb + j][1][sl];
            // SnowBucket: melt limited by snow store
            const float m   = fminf(fmaxf(Tk, 0.0f) * melt, s);
            const float sn  = s - m + ((Tk < 0.0f) ? Pk : 0.0f);
            // input gate + LinearBucket
            const float xin = (((Tk > 0.0f) ? Pk : 0.0f) + m) * gi;
            const float u   = xin + h;
            const float q   = u * gl;
            const float hn  = u - q;             // == h - q + xin
            qs[j] = red16(q * a);                // Q sum, broadcast to all lanes
            __builtin_nontemporal_store(hn, Hp + (size_t)j * ROW + hsOff);
            __builtin_nontemporal_store(sn, Sp + (size_t)j * ROW + hsOff);
            s = sn;
            h = hn;
        }
        if (b == 0) {                            // one EXEC toggle per 8 steps
#pragma unroll
            for (int j = 0; j < UNROLL; ++j)
                __builtin_nontemporal_store(qs[j], Qp + (size_t)j * NS + site);
        }
        Qp += (size_t)UNROLL * NS;
        Hp += (size_t)UNROLL * ROW;
        Sp += (size_t)UNROLL * ROW;
    };

    // Main loop: wait until the oldest 8 async loads (this body's group) are
    // done while the other group stays in flight; refill this group for t+16.
    int k = 0;
    for (; k + NBUF + UNROLL <= NT; k += UNROLL) {
        asm volatile("s_wait_asynccnt 0x8" ::: "memory");
        body(k);
        const int bb = k & 8;
#pragma unroll
        for (int j = 0; j < UNROLL; ++j) issue(k + NBUF + j, bb + j);
    }
    // Tail: 16 remaining staged steps (NT = 1096 = 8 * 137)
    asm volatile("s_wait_asynccnt 0x8" ::: "memory");
    body(k);
    k += UNROLL;
    asm volatile("s_wait_asynccnt 0x0" ::: "memory");
    body(k);
}

extern "C" void kernel_launch(void* const* d_in, const int* in_sizes, int n_in,
                              void* d_out, int out_size, void* d_ws, size_t ws_size,
                              hipStream_t stream) {
    (void)in_sizes; (void)n_in; (void)out_size; (void)d_ws; (void)ws_size;
    const float* P   = (const float*)d_in[0];
    const float* T   = (const float*)d_in[1];
    const float* w_i = (const float*)d_in[2];
    const float* w_o = (const float*)d_in[3];
    const float* w_l = (const float*)d_in[4];
    const float* w_s = (const float*)d_in[5];

    float* out = (float*)d_out;
    float* Q = out;                              // [NT, NS]
    float* H = Q + (size_t)NT * NS;              // [NT, NS, NH]
    float* S = H + (size_t)NT * NS * NH;         // [NT, NS, NH]

    const dim3 grid(NS / SPB);                   // 512 workgroups x 2 waves
    waternet_scan_kernel<<<grid, BLOCK, 0, stream>>>(P, T, w_i, w_o, w_l, w_s,
                                                     Q, H, S);
}